// AdaptiveCrpsKernelLoss_83270825935158
// MI455X (gfx1250) — compile-verified
//
#include <hip/hip_runtime.h>
#include <hip/hip_bf16.h>

// ---- problem constants (match reference) ----
#define LAMBDA_C 0.02f
#define K_C      3.3f
#define NB   32
#define NM   20
#define NHW  (96 * 96)            // 9216
#define NPIX (NB * NHW)           // 294912
#define NELEM ((float)NPIX * NM)  // 5898240
#define BLOCK 256
#define NV4   (NPIX / 4)          // 73728 float4 pixel-groups
#define GRID1 (NV4 / BLOCK)       // 288 blocks -> 288 partials in d_ws

typedef __attribute__((ext_vector_type(2))) float v2f;
typedef __attribute__((ext_vector_type(4))) float f4;
typedef __attribute__((ext_vector_type(8))) float v8f;

__device__ __forceinline__ f4 vabs4(f4 a) {
    f4 r;
    r.x = __builtin_fabsf(a.x); r.y = __builtin_fabsf(a.y);
    r.z = __builtin_fabsf(a.z); r.w = __builtin_fabsf(a.w);
    return r;
}
__device__ __forceinline__ f4 vrelu4(f4 a) {
    f4 r;
    r.x = fmaxf(a.x, 0.0f); r.y = fmaxf(a.y, 0.0f);
    r.z = fmaxf(a.z, 0.0f); r.w = fmaxf(a.w, 0.0f);
    return r;
}
__device__ __forceinline__ f4 vsqrt4(f4 a) {
    f4 r;
    r.x = __builtin_sqrtf(a.x); r.y = __builtin_sqrtf(a.y);
    r.z = __builtin_sqrtf(a.z); r.w = __builtin_sqrtf(a.w);
    return r;
}

// Full-precision 32-lane sum via V_WMMA_F32_16X16X4_F32.
// A (16x4 f32, 2 VGPRs): lane L<16 holds A[M=L][K=0] in a[0]; lane L>=16 holds
// A[M=L-16][K=2] in a[0]; a[1] (K=1/K=3) is zero. B = all ones (layout-free).
// => D[M][N] = v_M + v_{M+16} for every column N.
// Lane sums its 8 D-VGPRs: lanes 0-15 get sum(v_0..7 + v_16..23), lanes 16-31
// get sum(v_8..15 + v_24..31); one shfl_xor(16) completes the 32-lane total.
__device__ __forceinline__ float wave_sum32_wmma(float v) {
    v2f a; a[0] = v;    a[1] = 0.0f;
    v2f b; b[0] = 1.0f; b[1] = 1.0f;
    v8f c = {};
    v8f d = __builtin_amdgcn_wmma_f32_16x16x4_f32(
        /*neg_a=*/false, a, /*neg_b=*/false, b,
        /*c_mod=*/(short)0, c, /*reuse_a=*/false, /*reuse_b=*/false);
    float s = d[0] + d[1] + d[2] + d[3] + d[4] + d[5] + d[6] + d[7];
    s += __shfl_xor(s, 16, 32);
    return s;  // every lane holds the full 32-lane sum
}

__device__ __forceinline__ float block_sum(float v, float* lsum) {
    const int lane = threadIdx.x & 31;
    const int wid  = threadIdx.x >> 5;
    float ws = wave_sum32_wmma(v);
    if (lane == 0) lsum[wid] = ws;
    __syncthreads();
    float bs = 0.0f;
    if (threadIdx.x == 0) {
#pragma unroll
        for (int w = 0; w < BLOCK / 32; ++w) bs += lsum[w];
    }
    return bs;  // valid in thread 0 only
}

// Kernel 1: each thread processes 4 consecutive pixels (float4 lanes).
// 20 x global_load_b128 per thread, fully coalesced (member stride = 9216 f32,
// 16B-aligned). One deterministic partial per block -> d_ws.
__global__ void __launch_bounds__(BLOCK)
crps_partial_kernel(const float* __restrict__ fc, const float* __restrict__ tr,
                    float* __restrict__ part) {
    const int t   = blockIdx.x * BLOCK + threadIdx.x;  // [0, NV4)
    const int pix = t * 4;
    const int b   = pix / NHW;
    const int hw  = pix - b * NHW;
    const float* fbase = fc + (size_t)b * (NM * NHW) + hw;

    f4 x[NM];
#pragma unroll
    for (int i = 0; i < NM; ++i)
        x[i] = *(const f4*)(fbase + i * NHW);
    const f4 y = *(const f4*)(tr + pix);

    f4 sum = (f4)0.0f, sumsq = (f4)0.0f, sxy = (f4)0.0f, sxx = (f4)0.0f;
#pragma unroll
    for (int i = 0; i < NM; ++i) {
        sum   += x[i];
        sumsq += x[i] * x[i];
        sxy   += vabs4(x[i] - y);
    }
    // 190 unique pairs; diagonal is zero, off-diagonal counted twice in ref.
#pragma unroll
    for (int i = 0; i < NM; ++i) {
#pragma unroll
        for (int j = i + 1; j < NM; ++j)
            sxx += vabs4(x[i] - x[j]);
    }

    const f4 mean = sum * (1.0f / NM);
    f4 var = (sumsq - mean * sum) * (1.0f / (NM - 1));  // unbiased (ddof=1)
    var = vrelu4(var);                                  // guard fp negatives
    const f4 sd = vsqrt4(var);

    f4 pen = (f4)0.0f;
#pragma unroll
    for (int i = 0; i < NM; ++i)
        pen += vrelu4(vabs4(x[i] - mean) - K_C * sd);

    // crps = sxy/20 - 0.5*(2*sxx/400) = sxy/20 - sxx/400
    const f4 crps = sxy * (1.0f / NM) - sxx * (1.0f / (NM * NM));
    const f4 contrib = crps * (1.0f / NPIX) + pen * (LAMBDA_C / NELEM);
    const float v = contrib.x + contrib.y + contrib.z + contrib.w;

    __shared__ float lsum[BLOCK / 32];
    const float bs = block_sum(v, lsum);
    if (threadIdx.x == 0) part[blockIdx.x] = bs;
}

// Kernel 2: single block, fixed-order strided sum of the 288 partials, then
// the same WMMA wave reduction. Fully deterministic (no atomics).
__global__ void __launch_bounds__(BLOCK)
crps_final_kernel(const float* __restrict__ part, int n, float* __restrict__ out) {
    float s = 0.0f;
    for (int i = threadIdx.x; i < n; i += BLOCK) s += part[i];
    __shared__ float lsum[BLOCK / 32];
    const float bs = block_sum(s, lsum);
    if (threadIdx.x == 0) out[0] = bs;
}

extern "C" void kernel_launch(void* const* d_in, const int* in_sizes, int n_in,
                              void* d_out, int out_size, void* d_ws, size_t ws_size,
                              hipStream_t stream) {
    (void)in_sizes; (void)n_in; (void)out_size; (void)ws_size;
    const float* forecast = (const float*)d_in[0];  // [32,20,96,96] f32
    const float* truth    = (const float*)d_in[1];  // [32,96,96]   f32
    float* partials = (float*)d_ws;                 // GRID1 floats (1.1 KB)
    float* out      = (float*)d_out;                // 1 float

    crps_partial_kernel<<<GRID1, BLOCK, 0, stream>>>(forecast, truth, partials);
    crps_final_kernel<<<1, BLOCK, 0, stream>>>(partials, GRID1, out);
}